// CoPEModel_19997367730796
// MI455X (gfx1250) — compile-verified
//
#include <hip/hip_runtime.h>

typedef __attribute__((ext_vector_type(2))) float v2f;
typedef __attribute__((ext_vector_type(8))) float v8f;

#define FEATS 64
#define EPS 1e-5f

// ---------------- elementwise / setup kernels ----------------

__global__ void zero_kernel(float* __restrict__ p, int n) {
  int i = blockIdx.x * blockDim.x + threadIdx.x;
  if (i < n) p[i] = 0.0f;
}

__global__ void degree_kernel(const int* __restrict__ src, const int* __restrict__ dst,
                              float* __restrict__ deg_out, float* __restrict__ deg_in, int E) {
  int e = blockIdx.x * blockDim.x + threadIdx.x;
  if (e < E) {
    atomicAdd(&deg_out[src[e]], 1.0f);
    atomicAdd(&deg_in[dst[e]], 1.0f);
  }
}

// hs = x * rsqrt(max(deg_out,1)) ; also zero the SpMM accumulator
__global__ void scale_src_kernel(const float* __restrict__ x, const float* __restrict__ deg_out,
                                 float* __restrict__ hs, float* __restrict__ macc, int total) {
  int i = blockIdx.x * blockDim.x + threadIdx.x;
  if (i < total) {
    int row = i >> 6;
    hs[i] = x[i] * rsqrtf(fmaxf(deg_out[row], 1.0f));
    macc[i] = 0.0f;
  }
}

// macc[dst] += hs[src] : one wave per edge, each lane moves 2 floats
__global__ void spmm_kernel(const float* __restrict__ hs, const int* __restrict__ src,
                            const int* __restrict__ dst, float* __restrict__ macc, int nwork) {
  int tid = blockIdx.x * blockDim.x + threadIdx.x;
  if (tid < nwork) {
    int e = tid >> 5;
    int lane = tid & 31;
    int s = src[e];
    int d = dst[e];
    v2f v = *reinterpret_cast<const v2f*>(hs + s * FEATS + lane * 2);
    float* base = macc + d * FEATS + lane * 2;
    atomicAdd(base, v.x);
    atomicAdd(base + 1, v.y);
  }
}

// ---------------- WMMA GEMM with fused deg_in scale + bias + BN stats ----------------
// out[N x 64] = (macc * rsqrt(max(deg_in,1))) @ W + bias
// stats[0..63] += per-column sum, stats[64..127] += per-column sum of squares
// Grid: N/16 blocks, 128 threads (4 waves); wave w computes the 16x16 tile at
// rows rt*16..+15, cols w*16..+15, looping K in steps of 4 with v_wmma_f32_16x16x4_f32.
__global__ __launch_bounds__(128) void gemm_stats_kernel(
    const float* __restrict__ Am, const float* __restrict__ deg_in,
    const float* __restrict__ W, const float* __restrict__ bias,
    float* __restrict__ out, float* __restrict__ stats) {
  const int lane  = threadIdx.x & 31;
  const int wave  = threadIdx.x >> 5;      // column tile 0..3
  const int rt    = blockIdx.x;            // row tile
  const int m15   = lane & 15;
  const int khalf = (lane >> 4) << 1;      // lanes 0-15 -> K+0,K+1 ; lanes 16-31 -> K+2,K+3
  const int row   = rt * 16 + m15;         // A-matrix row for this lane
  const int col   = wave * 16 + m15;       // B/C/D column for this lane
  const float rs  = rsqrtf(fmaxf(deg_in[row], 1.0f));

  v8f acc = {};
#pragma unroll
  for (int kb = 0; kb < 16; ++kb) {
    const int k0 = kb * 4 + khalf;
    v2f a = *reinterpret_cast<const v2f*>(Am + row * FEATS + k0);
    a.x *= rs;
    a.y *= rs;
    v2f b;
    b.x = W[k0 * FEATS + col];
    b.y = W[(k0 + 1) * FEATS + col];
    acc = __builtin_amdgcn_wmma_f32_16x16x4_f32(
        /*neg_a=*/false, a, /*neg_b=*/false, b,
        /*c_mod=*/(short)0, acc, /*reuse_a=*/false, /*reuse_b=*/false);
  }

  const float bcol = bias[col];
  const int rowbase = rt * 16 + ((lane >> 4) << 3);  // C/D: lanes 16-31 hold M=8..15
  float s = 0.0f, q = 0.0f;
#pragma unroll
  for (int r = 0; r < 8; ++r) {
    float v = acc[r] + bcol;
    out[(rowbase + r) * FEATS + col] = v;
    s += v;
    q += v * v;
  }
  atomicAdd(&stats[col], s);
  atomicAdd(&stats[FEATS + col], q);
}

// stats -> per-column scale/shift:  scale = g*rsqrt(var+eps), shift = be - mu*scale
__global__ void bn_finalize_kernel(const float* __restrict__ stats,
                                   const float* __restrict__ g, const float* __restrict__ be,
                                   float* __restrict__ scsh, float invN) {
  int j = threadIdx.x;
  float mu  = stats[j] * invN;
  float var = stats[FEATS + j] * invN - mu * mu;
  float sc  = g[j] * rsqrtf(var + EPS);
  scsh[j] = sc;
  scsh[FEATS + j] = be[j] - mu * sc;
}

// hs = relu(h*scale+shift) * rsqrt(max(deg_out,1)) (layer-2 SpMM input); zero macc
__global__ void bn_relu_scale_kernel(const float* __restrict__ h, const float* __restrict__ scsh,
                                     const float* __restrict__ deg_out,
                                     float* __restrict__ hs, float* __restrict__ macc, int total) {
  int i = blockIdx.x * blockDim.x + threadIdx.x;
  if (i < total) {
    int col = i & 63;
    int row = i >> 6;
    float v = fmaxf(h[i] * scsh[col] + scsh[FEATS + col], 0.0f);
    hs[i] = v * rsqrtf(fmaxf(deg_out[row], 1.0f));
    macc[i] = 0.0f;
  }
}

// colsum[j] += sum over this block's 64 rows of relu(bn(h))[:, j]
__global__ __launch_bounds__(256) void bn_relu_colsum_kernel(
    const float* __restrict__ h, const float* __restrict__ scsh,
    float* __restrict__ colsum, int N) {
  const int col = threadIdx.x & 63;
  const int rg  = threadIdx.x >> 6;          // 0..3
  const int base = blockIdx.x * 64;
  const float sc = scsh[col];
  const float sh = scsh[FEATS + col];
  float acc = 0.0f;
#pragma unroll
  for (int i = 0; i < 16; ++i) {
    int row = base + rg + i * 4;
    if (row < N) acc += fmaxf(h[row * FEATS + col] * sc + sh, 0.0f);
  }
  atomicAdd(&colsum[col], acc);
}

// out[c] = sum_j (colsum[j]/N) * Wc[j][c] + bc[c]
__global__ void classifier_kernel(const float* __restrict__ colsum,
                                  const float* __restrict__ Wc, const float* __restrict__ bc,
                                  float* __restrict__ out, float invN) {
  __shared__ float p0[FEATS];
  __shared__ float p1[FEATS];
  int j = threadIdx.x;
  float m = colsum[j] * invN;
  p0[j] = m * Wc[j * 2 + 0];
  p1[j] = m * Wc[j * 2 + 1];
  __syncthreads();
  if (j == 0) {
    float a = 0.0f, b = 0.0f;
    for (int k = 0; k < FEATS; ++k) { a += p0[k]; b += p1[k]; }
    out[0] = a + bc[0];
    out[1] = b + bc[1];
  }
}

// ---------------- host-side launch sequence ----------------

extern "C" void kernel_launch(void* const* d_in, const int* in_sizes, int n_in,
                              void* d_out, int out_size, void* d_ws, size_t ws_size,
                              hipStream_t stream) {
  (void)n_in; (void)out_size; (void)ws_size;

  const float* x   = (const float*)d_in[0];
  const int*   src = (const int*)  d_in[1];
  const int*   dst = (const int*)  d_in[2];
  const float* W1  = (const float*)d_in[3];
  const float* b1  = (const float*)d_in[4];
  const float* g1  = (const float*)d_in[5];
  const float* be1 = (const float*)d_in[6];
  const float* W2  = (const float*)d_in[7];
  const float* b2  = (const float*)d_in[8];
  const float* g2  = (const float*)d_in[9];
  const float* be2 = (const float*)d_in[10];
  const float* Wc  = (const float*)d_in[11];
  const float* bc  = (const float*)d_in[12];
  float* out = (float*)d_out;

  const int N = in_sizes[0] / FEATS;  // 100000
  const int E = in_sizes[1];          // 1600000
  const int total = N * FEATS;
  const float invN = 1.0f / (float)N;

  // workspace layout (floats)
  float* ws = (float*)d_ws;
  float* deg_out = ws;                 // N
  float* deg_in  = deg_out + N;        // N
  float* stats1  = deg_in + N;         // 128
  float* scsh1   = stats1 + 128;       // 128
  float* stats2  = scsh1 + 128;        // 128
  float* scsh2   = stats2 + 128;       // 128
  float* colsum  = scsh2 + 128;        // 64
  float* bufA    = colsum + 64;        // N*64 (scaled features / SpMM input)
  float* bufB    = bufA + total;       // N*64 (SpMM accumulator)
  float* bufC    = bufB + total;       // N*64 (post-GEMM h)

  const int smallN = 2 * N + 576;      // degrees + stats + scsh + colsum

  const int B256 = 256;
  dim3 blk(B256);

  // 0) zero degrees + all small accumulators
  zero_kernel<<<(smallN + B256 - 1) / B256, blk, 0, stream>>>(ws, smallN);
  // 1) degrees
  degree_kernel<<<(E + B256 - 1) / B256, blk, 0, stream>>>(src, dst, deg_out, deg_in, E);
  // 2) layer 1: scale by rsqrt(deg_out), zero accumulator
  scale_src_kernel<<<(total + B256 - 1) / B256, blk, 0, stream>>>(x, deg_out, bufA, bufB, total);
  // 3) SpMM 1
  {
    int nwork = E * 32;
    spmm_kernel<<<(nwork + B256 - 1) / B256, blk, 0, stream>>>(bufA, src, dst, bufB, nwork);
  }
  // 4) GEMM 1 (fused rsqrt(deg_in), bias, BN stats) -- N assumed multiple of 16
  gemm_stats_kernel<<<N / 16, dim3(128), 0, stream>>>(bufB, deg_in, W1, b1, bufC, stats1);
  // 5) BN1 finalize
  bn_finalize_kernel<<<1, dim3(FEATS), 0, stream>>>(stats1, g1, be1, scsh1, invN);
  // 6) relu(bn1) * rsqrt(deg_out) -> layer-2 SpMM input; zero accumulator
  bn_relu_scale_kernel<<<(total + B256 - 1) / B256, blk, 0, stream>>>(bufC, scsh1, deg_out, bufA, bufB, total);
  // 7) SpMM 2
  {
    int nwork = E * 32;
    spmm_kernel<<<(nwork + B256 - 1) / B256, blk, 0, stream>>>(bufA, src, dst, bufB, nwork);
  }
  // 8) GEMM 2
  gemm_stats_kernel<<<N / 16, dim3(128), 0, stream>>>(bufB, deg_in, W2, b2, bufC, stats2);
  // 9) BN2 finalize
  bn_finalize_kernel<<<1, dim3(FEATS), 0, stream>>>(stats2, g2, be2, scsh2, invN);
  // 10) relu(bn2) column sums (graph mean readout, unscaled)
  bn_relu_colsum_kernel<<<(N + 63) / 64, blk, 0, stream>>>(bufC, scsh2, colsum, N);
  // 11) classifier: out = (colsum/N) @ Wc + bc
  classifier_kernel<<<1, dim3(FEATS), 0, stream>>>(colsum, Wc, bc, out, invN);
}